// MultiWordSelectionHead_17420387352655
// MI455X (gfx1250) — compile-verified
//
#include <hip/hip_runtime.h>
#include <hip/hip_bf16.h>
#include <cstddef>

// ---------------------------------------------------------------------------
// MultiWordSelectionHead for MI455X (gfx1250, wave32)
//   1) gather rows + dense GEMM (fp32 WMMA 16x16x4) + bias + LayerNorm -> d_ws
//   2) per-position candidate-embedding dot products -> logits
// ---------------------------------------------------------------------------

typedef float v2f __attribute__((ext_vector_type(2)));
typedef float v8f __attribute__((ext_vector_type(8)));

namespace {
constexpr int BN   = 32;            // batch
constexpr int SS   = 512;           // sequence length
constexpr int HH   = 768;           // hidden
constexpr int PP   = 76;            // masked positions per batch
constexpr int KCND = 64;            // candidates per position
constexpr int EE   = 768;           // embedding dim
constexpr int MROW = BN * PP;       // 2432 total rows
constexpr int LDSS = HH + 4;        // padded LDS row stride (772 dwords: bank-conflict-free column reads)
constexpr float LN_EPS = 1e-12f;
}

// ===========================================================================
// Kernel 1: 16 rows per block, 8 waves, each wave owns a 16x96 output slab.
//   lm[r, :] = LayerNorm(gather(seq, pos)[r, :] @ W + b) * gamma + beta
// ===========================================================================
__global__ void __launch_bounds__(256, 2)
gemm_ln_kernel(const float* __restrict__ seq,     // [B,S,H]
               const int*   __restrict__ pos,     // [B,P]
               const float* __restrict__ w,       // [H,E]
               const float* __restrict__ bias,    // [E]
               const float* __restrict__ gamma,   // [E]
               const float* __restrict__ beta,    // [E]
               float*       __restrict__ lm_out)  // [M,E]
{
    __shared__ float g[16 * LDSS];               // 48.25 KB: A tile, later lm tile

    const int row0 = blockIdx.x * 16;
    const int tid  = threadIdx.x;
    const int lane = tid & 31;
    const int wave = tid >> 5;

    // ---- stage 16 gathered activation rows into LDS (b128 loads/stores) ----
    for (int i = tid; i < 16 * (HH / 4); i += 256) {
        const int r  = i / (HH / 4);
        const int c4 = i % (HH / 4);
        const int mrow = row0 + r;
        const int b = mrow / PP;
        const int p = mrow - b * PP;
        const int s = pos[b * PP + p];
        const float4 v =
            ((const float4*)(seq + ((size_t)b * SS + (size_t)s) * HH))[c4];
        *(float4*)&g[r * LDSS + c4 * 4] = v;
    }
    __syncthreads();

    // ---- fp32 WMMA GEMM: D(16x96 per wave) += A(16x4) x B(4x96) over K=768 --
    const int n_base = wave * 96;                // 6 tiles of 16 columns
    const int mA     = lane & 15;                // A-fragment row
    const int khalf  = (lane >> 4) << 1;         // 0 (lanes 0-15) or 2 (lanes 16-31)

    v8f acc[6];
    for (int t = 0; t < 6; ++t) acc[t] = (v8f){0.f,0.f,0.f,0.f,0.f,0.f,0.f,0.f};

    #pragma unroll 2
    for (int k = 0; k < HH; k += 4) {
        // A fragment (shared across the 6 N-tiles): ds_load_b64
        const float* ap = &g[mA * LDSS + k + khalf];
        v2f a; a.x = ap[0]; a.y = ap[1];

        #pragma unroll
        for (int t = 0; t < 6; ++t) {
            const int n = n_base + t * 16;
            const float* bp = &w[(size_t)(k + khalf) * EE + n + mA];
            v2f bb; bb.x = bp[0]; bb.y = bp[EE];   // rows k+khalf, k+khalf+1
            acc[t] = __builtin_amdgcn_wmma_f32_16x16x4_f32(
                false, a, false, bb, (short)0, acc[t], false, false);
        }
    }

    // ---- spill biased accumulators back into the LDS tile for LayerNorm ----
    __syncthreads();                              // everyone done reading A
    {
        const int rhi = (lane >> 4) << 3;         // +8 for lanes 16-31
        #pragma unroll
        for (int t = 0; t < 6; ++t) {
            const int col = n_base + t * 16 + (lane & 15);
            const float bv = bias[col];
            #pragma unroll
            for (int j = 0; j < 8; ++j) {
                g[(j + rhi) * LDSS + col] = acc[t][j] + bv;
            }
        }
    }
    __syncthreads();

    // ---- LayerNorm: each wave normalizes 2 rows (wave32 shuffle reduce) ----
    for (int rr = 0; rr < 2; ++rr) {
        const int r = wave * 2 + rr;
        const float* rowp = &g[r * LDSS];
        float s = 0.f, s2 = 0.f;
        #pragma unroll
        for (int c = lane; c < EE; c += 32) {
            const float v = rowp[c];
            s += v; s2 += v * v;
        }
        #pragma unroll
        for (int off = 16; off > 0; off >>= 1) {
            s  += __shfl_xor(s,  off, 32);
            s2 += __shfl_xor(s2, off, 32);
        }
        const float mu  = s * (1.f / EE);
        const float var = s2 * (1.f / EE) - mu * mu;
        const float inv = rsqrtf(var + LN_EPS);
        float* outp = lm_out + (size_t)(row0 + r) * EE;
        #pragma unroll
        for (int c = lane; c < EE; c += 32) {
            outp[c] = (rowp[c] - mu) * inv * gamma[c] + beta[c];
        }
    }
}

// ===========================================================================
// Kernel 2: one block per (b,p). lm row cached in LDS; each of 8 waves dots
// 8 candidate embedding rows (float4 loads, wave32 xor-shuffle reduction).
// Per-row candidate matrices are unique per position -> not WMMA-shaped.
// ===========================================================================
__global__ void __launch_bounds__(256, 8)
logits_kernel(const float* __restrict__ lm,      // [M,E]
              const int*   __restrict__ cands,   // [B,P,K]
              const float* __restrict__ emb,     // [V,E]
              float*       __restrict__ out)     // [B,P,K]
{
    __shared__ float q[EE];                      // 3 KB query row

    const int bp   = blockIdx.x;                 // 0 .. M-1
    const int tid  = threadIdx.x;
    const int lane = tid & 31;
    const int wave = tid >> 5;

    for (int i = tid; i < EE / 4; i += 256) {
        *(float4*)&q[i * 4] = ((const float4*)(lm + (size_t)bp * EE))[i];
    }
    __syncthreads();

    const int* cs = cands + (size_t)bp * KCND;

    #pragma unroll
    for (int c = 0; c < 8; ++c) {
        const int k   = wave * 8 + c;
        const int vid = cs[k];
        const float4* ep = (const float4*)(emb + (size_t)vid * EE);

        if (c + 1 < 8) {                          // prefetch next candidate row
            const float* np = emb + (size_t)cs[k + 1] * EE;
            __builtin_prefetch(np + lane * 24, 0, 0);   // covers the 3 KB row
        }

        float dot = 0.f;
        #pragma unroll
        for (int i = lane; i < EE / 4; i += 32) { // 6 float4 per lane
            const float4 e4 = ep[i];
            const float4 q4 = *(const float4*)&q[i * 4];
            dot += e4.x * q4.x + e4.y * q4.y + e4.z * q4.z + e4.w * q4.w;
        }
        #pragma unroll
        for (int off = 16; off > 0; off >>= 1) dot += __shfl_xor(dot, off, 32);
        if (lane == 0) out[(size_t)bp * KCND + k] = dot;
    }
}

// ===========================================================================
extern "C" void kernel_launch(void* const* d_in, const int* in_sizes, int n_in,
                              void* d_out, int out_size, void* d_ws, size_t ws_size,
                              hipStream_t stream) {
    const float* seq   = (const float*)d_in[0];  // [B,S,H]
    const int*   pos   = (const int*)  d_in[1];  // [B,P]
    const int*   cands = (const int*)  d_in[2];  // [B,P,K]
    const float* emb   = (const float*)d_in[3];  // [V,E]
    const float* w     = (const float*)d_in[4];  // [H,E]
    const float* bias  = (const float*)d_in[5];  // [E]
    const float* gamma = (const float*)d_in[6];  // [E]
    const float* beta  = (const float*)d_in[7];  // [E]
    float* logits = (float*)d_out;               // [B,P,K]

    float* lm = (float*)d_ws;                    // [M,E] = 7.47 MB scratch

    gemm_ln_kernel<<<MROW / 16, 256, 0, stream>>>(seq, pos, w, bias, gamma, beta, lm);
    logits_kernel<<<MROW, 256, 0, stream>>>(lm, cands, emb, logits);
}